// Rpn2Proposal_52123723104377
// MI455X (gfx1250) — compile-verified
//
#include <hip/hip_runtime.h>
#include <hip/hip_bf16.h>
#include <stdint.h>

// ---------------- problem constants (from reference) ----------------
#define BATCH 8
#define NANCH 262144            // 2^18
#define KSEL  5000              // PERFORMANCE_COUNT
#define NOUT  1000              // PROPOSAL_COUNT
#define NEGV  (-1.0e9f)
#define MAXRATIO 4.1351666f     // |log(16/1000)|
#define NMS_THREADS 1024
#define ELEMS 5                 // ceil(KSEL / NMS_THREADS)

#ifndef __has_builtin
#define __has_builtin(x) 0
#endif

#if __has_builtin(__builtin_amdgcn_tensor_load_to_lds) && __has_builtin(__builtin_amdgcn_s_wait_tensorcnt)
#define USE_TDM 1
#else
#define USE_TDM 0
#endif

// LDS layout for the NMS kernel (dynamic LDS only -> offsets start at groupstaticsize()==0)
#define OFF_BOX    0u           // 4*KSEL floats  = 80000 B (winner-broadcast copy, TDM-filled)
#define OFF_SCORE  80000u       // KSEL floats    = 20000 B (staging for initial register load)
#define OFF_REDV   100000u      // 32 floats
#define OFF_REDI   100128u      // 32 ints
#define OFF_WIN    100256u      // winner idx + val
#define NMS_LDS_BYTES 100288u

// ---------------- TDM helpers (CDNA5 Tensor Data Mover) ----------------
#if USE_TDM
typedef unsigned int u32x4 __attribute__((ext_vector_type(4)));
typedef int          i32x8 __attribute__((ext_vector_type(8)));
typedef int          i32x4 __attribute__((ext_vector_type(4)));

// 1-D tile DMA: global -> LDS, nelem 4-byte elements (nelem <= 65535).
// D# per CDNA5 ISA: group0 {count=1, lds_addr, global_addr, type=2},
// group1 {data_size=4B, tensor_dim0=nelem, tensor_dim1=1, tile_dim0=nelem, tile_dim1=1,
//         tensor_dim0_stride=nelem}; groups 2/3 zero (2-D descriptor, dims 3/4 unused).
__device__ __forceinline__ void tdm_load_1d(unsigned lds_addr, const void* gptr, unsigned nelem) {
  unsigned long long ga = (unsigned long long)(uintptr_t)gptr;
  u32x4 g0;
  g0.x = 1u;                                                  // count=1, user descriptor
  g0.y = lds_addr;                                            // LDS byte address
  g0.z = (unsigned)(ga & 0xFFFFFFFFu);                        // global_addr[31:0]
  g0.w = (unsigned)((ga >> 32) & 0x01FFFFFFu) | (2u << 30);   // global_addr[56:32] | type=2
  i32x8 g1;
  g1[0] = (int)(2u << 16);                                    // data_size=2 (4 bytes), no flags
  g1[1] = (int)((nelem & 0xFFFFu) << 16);                     // tensor_dim0[15:0] @ bit48
  g1[2] = (int)((nelem >> 16) | (1u << 16));                  // tensor_dim0[31:16] | tensor_dim1=1
  g1[3] = (int)((nelem & 0xFFFFu) << 16);                     // tensor_dim1 hi=0 | tile_dim0=nelem
  g1[4] = 1;                                                  // tile_dim1=1, tile_dim2=0
  g1[5] = (int)nelem;                                         // tensor_dim0_stride[31:0]
  g1[6] = 0;                                                  // stride0[47:32] | stride1[15:0]
  g1[7] = 0;                                                  // stride1[47:16]
  i32x4 z4 = {0, 0, 0, 0};
#if defined(__clang_major__) && (__clang_major__ >= 23)
  i32x8 z8 = {0, 0, 0, 0, 0, 0, 0, 0};
  __builtin_amdgcn_tensor_load_to_lds(g0, g1, z4, z4, z8, 0);
#else
  __builtin_amdgcn_tensor_load_to_lds(g0, g1, z4, z4, 0);
#endif
}
#endif // USE_TDM

// ---------------- kernel 0: zero the per-batch histograms ----------------
__global__ void init_kernel(unsigned* hist) {
  int i = blockIdx.x * blockDim.x + threadIdx.x;
  if (i < BATCH * 2048) hist[i] = 0u;
}

// ---------------- kernel 1: masked score -> sortable key + 2048-bin histogram ----------------
__global__ void key_hist_kernel(const float* __restrict__ rpn_score,
                                const float4* __restrict__ anchors,
                                unsigned* __restrict__ keys,
                                unsigned* __restrict__ hist) {
  __shared__ unsigned lh[2048];
  for (int j = threadIdx.x; j < 2048; j += blockDim.x) lh[j] = 0u;
  __syncthreads();

  int gid = blockIdx.x * blockDim.x + threadIdx.x;   // [0, BATCH*NANCH)
  int b = gid >> 18;
  int i = gid & (NANCH - 1);
  float4 a = anchors[i];
  bool valid = (a.z <= 1.0f) && (a.w <= 1.0f) && (a.x >= 0.0f) && (a.y >= 0.0f);
  float s = valid ? rpn_score[gid] : -1.0f;
  // order-preserving float -> uint key (descending = larger key)
  unsigned fb = __float_as_uint(s);
  unsigned k = (fb & 0x80000000u) ? ~fb : (fb | 0x80000000u);
  keys[gid] = k;
  atomicAdd(&lh[k >> 21], 1u);
  __syncthreads();

  unsigned* gh = hist + (unsigned)b * 2048u;  // block is entirely inside one batch (256 | NANCH)
  for (int j = threadIdx.x; j < 2048; j += blockDim.x)
    if (lh[j]) atomicAdd(&gh[j], lh[j]);
}

// ---------------- kernel 2: find bucket of the 5000th score; init counters ----------------
__global__ void threshold_kernel(const unsigned* __restrict__ hist, unsigned* __restrict__ meta) {
  int b = threadIdx.x;
  if (b >= BATCH) return;
  const unsigned* h = hist + b * 2048;
  unsigned cum = 0; int T = 0; unsigned need = KSEL;
  for (int bin = 2047; bin >= 0; --bin) {
    unsigned c = h[bin];
    if (cum + c >= (unsigned)KSEL) { T = bin; need = (unsigned)KSEL - cum; break; }
    cum += c; T = bin; need = (unsigned)KSEL - cum;
  }
  unsigned* m = meta + b * 4;
  m[0] = (unsigned)T;  // boundary bucket
  m[1] = need;         // how many to take from boundary bucket
  m[2] = 0u;           // output count
  m[3] = 0u;           // boundary-bucket claim count
}

// ---------------- kernel 3: compact top-KSEL; decode + clip boxes only for winners ----------------
__global__ void compact_kernel(const unsigned* __restrict__ keys,
                               const float4* __restrict__ rpn_regress,
                               const float4* __restrict__ anchors,
                               unsigned* __restrict__ meta,
                               float* __restrict__ selScores,
                               float4* __restrict__ selBoxes) {
  int gid = blockIdx.x * blockDim.x + threadIdx.x;
  int b = gid >> 18;
  int i = gid & (NANCH - 1);
  unsigned k = keys[gid];
  unsigned bucket = k >> 21;
  unsigned* m = meta + b * 4;
  unsigned T = m[0];
  bool take = false;
  if (bucket > T) {
    take = true;
  } else if (bucket == T) {
    unsigned r = atomicAdd(&m[3], 1u);
    take = (r < m[1]);
  }
  if (!take) return;
  unsigned pos = atomicAdd(&m[2], 1u);
  if (pos >= (unsigned)KSEL) return;  // safety

  // invert key -> masked score
  unsigned fb = (k & 0x80000000u) ? (k & 0x7FFFFFFFu) : ~k;
  float score = __uint_as_float(fb);

  // delta2bbox (MEAN=0, STD={.1,.1,.2,.2}) + clip to [0,1]
  float4 a = anchors[i];
  float4 d = rpn_regress[gid];
  float dx = d.x * 0.1f, dy = d.y * 0.1f;
  float dw = fminf(fmaxf(d.z * 0.2f, -MAXRATIO), MAXRATIO);
  float dh = fminf(fmaxf(d.w * 0.2f, -MAXRATIO), MAXRATIO);
  float aw = a.z - a.x, ah = a.w - a.y;
  float acx = (a.z + a.x) * 0.5f, acy = (a.w + a.y) * 0.5f;
  float cx = acx + dx * aw, cy = acy + dy * ah;
  float w = aw * __expf(dw), h = ah * __expf(dh);
  float4 ob;
  ob.x = fminf(fmaxf(cx - 0.5f * w, 0.0f), 1.0f);
  ob.y = fminf(fmaxf(cy - 0.5f * h, 0.0f), 1.0f);
  ob.z = fminf(fmaxf(cx + 0.5f * w, 0.0f), 1.0f);
  ob.w = fminf(fmaxf(cy + 0.5f * h, 0.0f), 1.0f);

  selBoxes[(size_t)b * KSEL + pos] = ob;
  selScores[(size_t)b * KSEL + pos] = score;
}

// ---------------- kernel 4: sequential soft-NMS, register-resident scores ----------------
// One workgroup per batch; 1024 threads = 32 wave32s. Boxes live in LDS (TDM-filled,
// used only for winner broadcast); each thread keeps its <=5 scores/boxes/areas in
// registers. 2 barriers per round; per-round LDS traffic = 1 broadcast b128 read.
__global__ __launch_bounds__(NMS_THREADS) void nms_kernel(const float* __restrict__ selScores,
                                                          const float4* __restrict__ selBoxes,
                                                          float* __restrict__ out) {
  extern __shared__ char smem[];
  float4* sBox4 = (float4*)(smem + OFF_BOX);
  float*  sScore = (float*)(smem + OFF_SCORE);
  float*  sRedV = (float*)(smem + OFF_REDV);
  int*    sRedI = (int*)  (smem + OFF_REDI);
  int*    sWinI = (int*)  (smem + OFF_WIN);
  float*  sWinV = (float*)(smem + OFF_WIN + 4);

  const int b   = blockIdx.x;
  const int tid = threadIdx.x;
  const float4* gBox = selBoxes + (size_t)b * KSEL;
  const float*  gScr = selScores + (size_t)b * KSEL;

#if USE_TDM
  if (tid < 32) {  // wave 0 issues both DMA descriptors
    unsigned base = __builtin_amdgcn_groupstaticsize();  // dynamic LDS starts here (==0)
    tdm_load_1d(base + OFF_BOX,   gBox, 4u * KSEL);      // 80000 B tile
    tdm_load_1d(base + OFF_SCORE, gScr, (unsigned)KSEL); // 20000 B tile
    __builtin_amdgcn_s_wait_tensorcnt((short)0);
  }
#else
  for (int j = tid; j < KSEL; j += NMS_THREADS) {
    sBox4[j] = gBox[j];
    sScore[j] = gScr[j];
  }
#endif
  __syncthreads();

  // pull this thread's elements into registers
  float sreg[ELEMS], rx1[ELEMS], ry1[ELEMS], rx2[ELEMS], ry2[ELEMS], rar[ELEMS];
#pragma unroll
  for (int k = 0; k < ELEMS; ++k) {
    int j = tid + k * NMS_THREADS;
    if (j < KSEL) {
      float4 bb = sBox4[j];
      rx1[k] = bb.x; ry1[k] = bb.y; rx2[k] = bb.z; ry2[k] = bb.w;
      rar[k] = (bb.z - bb.x) * (bb.w - bb.y);
      sreg[k] = sScore[j];
    } else {
      rx1[k] = ry1[k] = rx2[k] = ry2[k] = rar[k] = 0.0f;
      sreg[k] = -3.0e38f;
    }
  }

  // initial local argmax (ascending j -> keeps first max, matching jnp.argmax ties)
  float bv = -3.0e38f; int bi = 0x7fffffff;
#pragma unroll
  for (int k = 0; k < ELEMS; ++k) {
    int j = tid + k * NMS_THREADS;
    if (j < KSEL && sreg[k] > bv) { bv = sreg[k]; bi = j; }
  }

  const int lane = tid & 31, wid = tid >> 5;

  for (int it = 0; it < NOUT; ++it) {
    // --- global argmax: wave32 shuffle reduce, tie-break on smaller index ---
    float rv = bv; int ri = bi;
    for (int off = 16; off > 0; off >>= 1) {
      float ov = __shfl_down(rv, off);
      int   oi = __shfl_down(ri, off);
      if (ov > rv || (ov == rv && oi < ri)) { rv = ov; ri = oi; }
    }
    if (lane == 0) { sRedV[wid] = rv; sRedI[wid] = ri; }
    __syncthreads();                               // B1
    if (tid < 32) {
      rv = sRedV[lane]; ri = sRedI[lane];
      for (int off = 16; off > 0; off >>= 1) {
        float ov = __shfl_down(rv, off);
        int   oi = __shfl_down(ri, off);
        if (ov > rv || (ov == rv && oi < ri)) { rv = ov; ri = oi; }
      }
      if (lane == 0) { sWinI[0] = ri; sWinV[0] = rv; }
    }
    __syncthreads();                               // B2

    const int   wi = sWinI[0];
    const float wv = sWinV[0];
    const float4 wb = sBox4[wi];                   // LDS broadcast read
    const float wa = (wb.z - wb.x) * (wb.w - wb.y); // bit-identical to per-elem area calc

    // --- emit selected box (pre-update score semantics, as in the reference) ---
    if (tid == 0) {
      float* o = out + ((size_t)b * NOUT + it) * 4;
      bool ok = wv > NEGV * 0.5f;
      o[0] = ok ? wb.x : 0.0f; o[1] = ok ? wb.y : 0.0f;
      o[2] = ok ? wb.z : 0.0f; o[3] = ok ? wb.w : 0.0f;
    }

    // --- register-only soft-NMS decay, fused with next round's local argmax ---
    bv = -3.0e38f; bi = 0x7fffffff;
#pragma unroll
    for (int k = 0; k < ELEMS; ++k) {
      int j = tid + k * NMS_THREADS;
      float s = sreg[k];
      float iw = fminf(wb.z, rx2[k]) - fmaxf(wb.x, rx1[k]);
      float ih = fminf(wb.w, ry2[k]) - fmaxf(wb.y, ry1[k]);
      float inter = fmaxf(iw, 0.0f) * fmaxf(ih, 0.0f);
      float iou = inter / (wa + rar[k] - inter + 1e-12f);
      s = s * __expf(-iou * iou);                  // exp(-0.5*iou^2/SIGMA), SIGMA=0.5
      if (j == wi) s = NEGV;
      sreg[k] = s;
      if (j < KSEL && s > bv) { bv = s; bi = j; }
    }
    // no barrier needed: round n+1 sRed writes happen after B2(n); wave-0 sRed reads
    // happen strictly between B1 and B2 of the same round.
  }
}

// ---------------- launch ----------------
extern "C" void kernel_launch(void* const* d_in, const int* in_sizes, int n_in,
                              void* d_out, int out_size, void* d_ws, size_t ws_size,
                              hipStream_t stream) {
  const float*  rpn_score   = (const float*)d_in[0];    // (B, N, 1)
  const float4* rpn_regress = (const float4*)d_in[1];   // (B, N, 4)
  const float4* anchors     = (const float4*)d_in[2];   // (N, 4)
  float* out = (float*)d_out;                           // (B, 1000, 4)

  // workspace layout (~8.9 MB)
  char* ws = (char*)d_ws;
  unsigned* keys      = (unsigned*)ws;                          // B*N*4      = 8388608 B
  unsigned* hist      = (unsigned*)(ws + 8388608);              // B*2048*4   = 65536 B
  unsigned* meta      = (unsigned*)(ws + 8388608 + 65536);      // B*4*4      = 128 B
  float*    selScores = (float*)(ws + 8388608 + 65536 + 256);   // B*KSEL*4   = 160000 B
  float4*   selBoxes  = (float4*)(ws + 8388608 + 65536 + 256 + 160000); // B*KSEL*16 = 640000 B

  const int total = BATCH * NANCH;
  init_kernel<<<(BATCH * 2048 + 255) / 256, 256, 0, stream>>>(hist);
  key_hist_kernel<<<total / 256, 256, 0, stream>>>(rpn_score, anchors, keys, hist);
  threshold_kernel<<<1, 32, 0, stream>>>(hist, meta);
  compact_kernel<<<total / 256, 256, 0, stream>>>(keys, rpn_regress, anchors, meta, selScores, selBoxes);
  nms_kernel<<<BATCH, NMS_THREADS, NMS_LDS_BYTES, stream>>>(selScores, selBoxes, out);
}